// GraphNet_22969485099881
// MI455X (gfx1250) — compile-verified
//
#include <hip/hip_runtime.h>
#include <math.h>

#define N_NODES 50000
#define N_EDGES 500000
#define LN_EPS  1e-6f

typedef float v2f __attribute__((ext_vector_type(2)));
typedef float v8f __attribute__((ext_vector_type(8)));

// gfx1250 has V_TANH_F32 (hardware transcendental); use it if clang exposes it.
__device__ __forceinline__ float fast_tanh(float x) {
#if __has_builtin(__builtin_amdgcn_tanhf)
  return __builtin_amdgcn_tanhf(x);
#else
  return tanhf(x);
#endif
}

// ---------------------------------------------------------------------------
// WMMA MLP layer: for R row-tiles of 16 rows each,
//   Y[t,16x16] = act(X[t,16xK] @ W[K x ldw][:, colbase:+16] + bias)
// One B load per k-step is reused across all R row tiles (R independent
// accumulators -> deep WMMA pipelining, R-fold cut in weight traffic).
// A layout (f32 16x4): lanes 0-15 row M=lane hold K={k,k+1}; lanes 16-31 K={k+2,k+3}
// B layout (f32 4x16): V0/V1 lanes 0-15 = rows k,k+1 ; lanes 16-31 = rows k+2,k+3
// C/D layout: VGPR r, lanes 0-15 -> M=r ; lanes 16-31 -> M=r+8 ; N = lane&15
// ---------------------------------------------------------------------------
template <int R>
__device__ __forceinline__ void wmma_mlp_layer(const float* __restrict__ X, int ldx,
                                               const float* __restrict__ W, int ldw,
                                               int colbase, int K, float* Y, int ldy,
                                               const float* bias, bool act, int lane) {
  const int idx = lane & 15;
  const int hf  = lane >> 4;
  v8f acc[R];
#pragma unroll
  for (int t = 0; t < R; ++t) acc[t] = (v8f){};
  for (int k = 0; k < K; k += 4) {
    const int ka = k + 2 * hf;
    v2f b;
    b[0] = W[ka * ldw + colbase + idx];
    b[1] = W[(ka + 1) * ldw + colbase + idx];
#pragma unroll
    for (int t = 0; t < R; ++t) {
      v2f a;
      a[0] = X[(t * 16 + idx) * ldx + ka];
      a[1] = X[(t * 16 + idx) * ldx + ka + 1];
      acc[t] = __builtin_amdgcn_wmma_f32_16x16x4_f32(false, a, false, b, (short)0,
                                                     acc[t], false, false);
    }
  }
  const float bv = bias[colbase + idx];
#pragma unroll
  for (int t = 0; t < R; ++t) {
#pragma unroll
    for (int r = 0; r < 8; ++r) {
      float v = acc[t][r] + bv;
      if (act) v = fast_tanh(v);
      Y[(t * 16 + r + 8 * hf) * ldy + colbase + idx] = v;
    }
  }
}

// orderable-uint encoding of float for atomicMax-based segment max
__device__ __forceinline__ unsigned enc_f32(float f) {
  unsigned u = __float_as_uint(f);
  return (u >> 31) ? ~u : (u | 0x80000000u);
}
__device__ __forceinline__ float dec_f32(unsigned u) {
  unsigned b = (u & 0x80000000u) ? (u & 0x7FFFFFFFu) : ~u;
  return __uint_as_float(b);
}

// ---------------------------------------------------------------------------
// misc small kernels
// ---------------------------------------------------------------------------
__global__ __launch_bounds__(256) void fill_zero_kernel(float* p, long n) {
  long i = (long)blockIdx.x * 256 + threadIdx.x;
  if (i < n) p[i] = 0.f;
}

__device__ __forceinline__ void ln8(const float* x, const float* s, const float* b,
                                    float* out) {
  float m = 0.f;
  for (int i = 0; i < 8; ++i) m += x[i];
  m *= 0.125f;
  float v = 0.f;
  for (int i = 0; i < 8; ++i) { float d = x[i] - m; v += d * d; }
  v *= 0.125f;
  float rs = rsqrtf(v + LN_EPS);
  for (int i = 0; i < 8; ++i) out[i] = (x[i] - m) * rs * s[i] + b[i];
}

// globals: embed, then g1 = LN0(2*g0), g2 = LN1(2*g1). gbuf = {g0, g1}, g_out = g2
__global__ void glob_kernel(const float* g_in, const float* w, const float* b,
                            const float* ln_s0, const float* ln_b0,
                            const float* ln_s1, const float* ln_b1,
                            float* gbuf, float* g_out) {
  if (threadIdx.x != 0) return;
  float g0[8], t[8], g1[8], g2[8];
  for (int n = 0; n < 8; ++n) {
    float s = b[n];
    for (int k = 0; k < 8; ++k) s += g_in[k] * w[k * 8 + n];
    g0[n] = s;
    gbuf[n] = s;
  }
  for (int i = 0; i < 8; ++i) t[i] = 2.f * g0[i];
  ln8(t, ln_s0, ln_b0, g1);
  for (int i = 0; i < 8; ++i) gbuf[8 + i] = g1[i];
  for (int i = 0; i < 8; ++i) t[i] = 2.f * g1[i];
  ln8(t, ln_s1, ln_b1, g2);
  for (int i = 0; i < 8; ++i) g_out[i] = g2[i];
}

// Y[16 rows tile,16] = X[.,16] @ W[16,16] + b   (embed_node / dec_node)
__global__ __launch_bounds__(32) void gemm_16x16_kernel(const float* X, const float* W,
                                                        const float* bias, float* Y) {
  const int lane = threadIdx.x & 31, idx = lane & 15, hf = lane >> 4;
  const long base = (long)blockIdx.x * 16;
  v8f acc = {};
#pragma unroll
  for (int k = 0; k < 16; k += 4) {
    const int ka = k + 2 * hf;
    v2f a, b;
    a[0] = X[(base + idx) * 16 + ka];
    a[1] = X[(base + idx) * 16 + ka + 1];
    b[0] = W[ka * 16 + idx];
    b[1] = W[(ka + 1) * 16 + idx];
    acc = __builtin_amdgcn_wmma_f32_16x16x4_f32(false, a, false, b, (short)0, acc,
                                                false, false);
  }
  const float bv = bias[idx];
#pragma unroll
  for (int r = 0; r < 8; ++r) Y[(base + r + 8 * hf) * 16 + idx] = acc[r] + bv;
}

// Y[16 rows tile,128] = X[.,16] @ W[16,128] + b   (embed_edge)
__global__ __launch_bounds__(256) void embed_edge_kernel(const float* X, const float* W,
                                                         const float* bias, float* Y) {
  const int tid = threadIdx.x, wave = tid >> 5, lane = tid & 31;
  const int idx = lane & 15, hf = lane >> 4;
  const long base = (long)blockIdx.x * 16;
  const int colbase = wave * 16;
  v8f acc = {};
#pragma unroll
  for (int k = 0; k < 16; k += 4) {
    const int ka = k + 2 * hf;
    v2f a, b;
    a[0] = X[(base + idx) * 16 + ka];
    a[1] = X[(base + idx) * 16 + ka + 1];
    b[0] = W[ka * 128 + colbase + idx];
    b[1] = W[(ka + 1) * 128 + colbase + idx];
    acc = __builtin_amdgcn_wmma_f32_16x16x4_f32(false, a, false, b, (short)0, acc,
                                                false, false);
  }
  const float bv = bias[colbase + idx];
#pragma unroll
  for (int r = 0; r < 8; ++r)
    Y[(base + r + 8 * hf) * 128 + colbase + idx] = acc[r] + bv;
}

// Y[16 rows tile,64] = X[.,128] @ W[128,64] + b   (dec_edge)
__global__ __launch_bounds__(128) void dec_edge_kernel(const float* X, const float* W,
                                                       const float* bias, float* Y) {
  const int tid = threadIdx.x, wave = tid >> 5, lane = tid & 31;
  const int idx = lane & 15, hf = lane >> 4;
  const long base = (long)blockIdx.x * 16;
  const int colbase = wave * 16;
  v8f acc = {};
  for (int k = 0; k < 128; k += 4) {
    const int ka = k + 2 * hf;
    v2f a, b;
    a[0] = X[(base + idx) * 128 + ka];
    a[1] = X[(base + idx) * 128 + ka + 1];
    b[0] = W[ka * 64 + colbase + idx];
    b[1] = W[(ka + 1) * 64 + colbase + idx];
    acc = __builtin_amdgcn_wmma_f32_16x16x4_f32(false, a, false, b, (short)0, acc,
                                                false, false);
  }
  const float bv = bias[colbase + idx];
#pragma unroll
  for (int r = 0; r < 8; ++r)
    Y[(base + r + 8 * hf) * 64 + colbase + idx] = acc[r] + bv;
}

// ---------------------------------------------------------------------------
// Edge MLP + attention-logit kernel: one 32-edge tile (2 WMMA row tiles) per
// 256-thread block; 8 waves each own a 16-col slice, reuse B across row tiles.
// Static LDS: 32*168*4 + 2*32*128*4 = 54272 B.
// ---------------------------------------------------------------------------
struct EdgeParams {
  const float* edges_cur; const float* nodes_cur; const float* glob;
  const int* senders; const int* receivers;
  const float *ew0, *eb0, *ew1, *eb1, *ew2, *eb2, *ew3, *eb3;
  const float *aw0, *ab0, *aw1, *ab1, *aw2, *ab2;
  float* new_edges; float* logits; unsigned* maxenc;
};

__global__ __launch_bounds__(256) void edge_mlp_kernel(EdgeParams p) {
  __shared__ float A[32 * 168];
  __shared__ float B[32 * 128];
  __shared__ float C[32 * 128];
  __shared__ int sidx[32], ridx[32];
  const int tid = threadIdx.x, wave = tid >> 5, lane = tid & 31;
  const long base = (long)blockIdx.x * 32;
  const int colbase = wave * 16;

  if (tid < 32) {
    sidx[tid] = p.senders[base + tid];
    ridx[tid] = p.receivers[base + tid];
  }
  __syncthreads();
  // A = concat(edges[128], nodes[snd][16], nodes[rcv][16], glob[8])
  for (int i = tid; i < 32 * 168; i += 256) {
    const int r = i / 168, c = i % 168;
    float v;
    if (c < 128)      v = p.edges_cur[(base + r) * 128 + c];
    else if (c < 144) v = p.nodes_cur[(long)sidx[r] * 16 + (c - 128)];
    else if (c < 160) v = p.nodes_cur[(long)ridx[r] * 16 + (c - 144)];
    else              v = p.glob[c - 160];
    A[r * 168 + c] = v;
  }
  __syncthreads();

  wmma_mlp_layer<2>(A, 168, p.ew0, 128, colbase, 168, B, 128, p.eb0, true, lane);
  __syncthreads();
  wmma_mlp_layer<2>(B, 128, p.ew1, 128, colbase, 128, C, 128, p.eb1, true, lane);
  __syncthreads();
  wmma_mlp_layer<2>(C, 128, p.ew2, 128, colbase, 128, B, 128, p.eb2, true, lane);
  __syncthreads();
  wmma_mlp_layer<2>(B, 128, p.ew3, 128, colbase, 128, C, 128, p.eb3, true, lane);
  __syncthreads();  // C = new_edges (pre-attention)

  // write new_edges to global; splice into A[:, :128] for the attention MLP
  for (int i = tid; i < 32 * 128; i += 256) {
    const int r = i >> 7, c = i & 127;
    const float v = C[i];
    p.new_edges[(base + r) * 128 + c] = v;
    A[r * 168 + c] = v;
  }
  __syncthreads();

  wmma_mlp_layer<2>(A, 168, p.aw0, 128, colbase, 168, B, 128, p.ab0, true, lane);
  __syncthreads();
  wmma_mlp_layer<2>(B, 128, p.aw1, 128, colbase, 128, C, 128, p.ab1, true, lane);
  __syncthreads();

  // final 128 -> 1 logit per edge; each wave handles 4 edges
#pragma unroll
  for (int q = 0; q < 4; ++q) {
    const int rr = wave * 4 + q;
    float part = 0.f;
#pragma unroll
    for (int j = 0; j < 4; ++j) {
      const int k = lane + 32 * j;
      part += C[rr * 128 + k] * p.aw2[k];
    }
    for (int m = 16; m >= 1; m >>= 1) part += __shfl_xor(part, m, 32);
    if (lane == 0) {
      const float lg = fast_tanh(part + p.ab2[0]);
      p.logits[base + rr] = lg;
      atomicMax(&p.maxenc[ridx[rr]], enc_f32(lg));
    }
  }
}

// ---------------------------------------------------------------------------
// segment softmax helpers
// ---------------------------------------------------------------------------
__global__ __launch_bounds__(256) void denom_kernel(const float* logits,
                                                    const int* receivers,
                                                    const unsigned* maxenc,
                                                    float* denom) {
  const int e = blockIdx.x * 256 + threadIdx.x;
  if (e < N_EDGES) {
    const int r = receivers[e];
    atomicAdd(&denom[r], expf(logits[e] - dec_f32(maxenc[r])));
  }
}

__global__ __launch_bounds__(256) void att_kernel(const float* logits,
                                                  const int* receivers,
                                                  const unsigned* maxenc,
                                                  const float* denom, float* att) {
  const int e = blockIdx.x * 256 + threadIdx.x;
  if (e < N_EDGES) {
    const int r = receivers[e];
    att[e] = expf(logits[e] - dec_f32(maxenc[r])) / denom[r];
  }
}

// scale new_edges by att in-place + scatter-add into sent/recv aggregates.
// one thread = one edge x 4 features  (grid = N_EDGES*32/256)
__global__ __launch_bounds__(256) void scatter_kernel(float* ne, const float* att,
                                                      const int* senders,
                                                      const int* receivers,
                                                      float* sent, float* recv) {
  const long gid = (long)blockIdx.x * 256 + threadIdx.x;
  const long e = gid >> 5;
  const int c0 = (threadIdx.x & 31) * 4;
  const float a = att[e];
  const long s = senders[e];
  const long r = receivers[e];
#pragma unroll
  for (int j = 0; j < 4; ++j) {
    const float v = ne[e * 128 + c0 + j] * a;
    ne[e * 128 + c0 + j] = v;
    atomicAdd(&sent[s * 128 + c0 + j], v);
    atomicAdd(&recv[r * 128 + c0 + j], v);
  }
}

// ---------------------------------------------------------------------------
// Node MLP + skip + LayerNorm(16): one 16-node tile per 256-thread block
// ---------------------------------------------------------------------------
struct NodeParams {
  const float* nodes_cur; const float* sent; const float* recv; const float* glob;
  const float *nw0, *nb0, *nw1, *nb1, *nw2, *nb2;
  const float* ln_s; const float* ln_b;
  float* nodes_next;
};

__global__ __launch_bounds__(256) void node_mlp_kernel(NodeParams p) {
  __shared__ float A[16 * 280];
  __shared__ float B[16 * 128];
  __shared__ float C[16 * 128];
  __shared__ float Y[16 * 16];
  const int tid = threadIdx.x, wave = tid >> 5, lane = tid & 31;
  const long base = (long)blockIdx.x * 16;
  const int colbase = wave * 16;

  for (int i = tid; i < 16 * 280; i += 256) {
    const int r = i / 280, c = i % 280;
    float v;
    if (c < 16)       v = p.nodes_cur[(base + r) * 16 + c];
    else if (c < 144) v = p.sent[(base + r) * 128 + (c - 16)];
    else if (c < 272) v = p.recv[(base + r) * 128 + (c - 144)];
    else              v = p.glob[c - 272];
    A[r * 280 + c] = v;
  }
  __syncthreads();

  wmma_mlp_layer<1>(A, 280, p.nw0, 128, colbase, 280, B, 128, p.nb0, true, lane);
  __syncthreads();
  wmma_mlp_layer<1>(B, 128, p.nw1, 128, colbase, 128, C, 128, p.nb1, true, lane);
  __syncthreads();
  if (wave == 0) {  // wave-uniform branch: EXEC stays all-ones inside
    wmma_mlp_layer<1>(C, 128, p.nw2, 16, 0, 128, Y, 16, p.nb2, true, lane);
  }
  __syncthreads();

  if (tid < 16) {  // skip + LN over 16 features, one row per thread
    const int r = tid;
    float x[16];
    float mean = 0.f;
    for (int c = 0; c < 16; ++c) {
      x[c] = Y[r * 16 + c] + p.nodes_cur[(base + r) * 16 + c];
      mean += x[c];
    }
    mean *= (1.f / 16.f);
    float var = 0.f;
    for (int c = 0; c < 16; ++c) { const float d = x[c] - mean; var += d * d; }
    var *= (1.f / 16.f);
    const float rs = rsqrtf(var + LN_EPS);
    for (int c = 0; c < 16; ++c)
      p.nodes_next[(base + r) * 16 + c] = (x[c] - mean) * rs * p.ln_s[c] + p.ln_b[c];
  }
}

// edges = LN(ne + edges_old) over 128 feats; one wave32 per edge, result in ne
__global__ __launch_bounds__(256) void edge_ln_kernel(float* ne, const float* eo,
                                                      const float* s, const float* b) {
  const long gid = (long)blockIdx.x * 256 + threadIdx.x;
  const long e = gid >> 5;
  const int lane = threadIdx.x & 31;
  float x[4];
  float sum = 0.f;
#pragma unroll
  for (int j = 0; j < 4; ++j) {
    const int c = lane + 32 * j;
    x[j] = ne[e * 128 + c] + eo[e * 128 + c];
    sum += x[j];
  }
  for (int m = 16; m >= 1; m >>= 1) sum += __shfl_xor(sum, m, 32);
  const float mean = sum * (1.f / 128.f);
  float var = 0.f;
#pragma unroll
  for (int j = 0; j < 4; ++j) { const float d = x[j] - mean; var += d * d; }
  for (int m = 16; m >= 1; m >>= 1) var += __shfl_xor(var, m, 32);
  var *= (1.f / 128.f);
  const float rs = rsqrtf(var + LN_EPS);
#pragma unroll
  for (int j = 0; j < 4; ++j) {
    const int c = lane + 32 * j;
    ne[e * 128 + c] = (x[j] - mean) * rs * s[c] + b[c];
  }
}

// ---------------------------------------------------------------------------
// host driver
// ---------------------------------------------------------------------------
extern "C" void kernel_launch(void* const* d_in, const int* in_sizes, int n_in,
                              void* d_out, int out_size, void* d_ws, size_t ws_size,
                              hipStream_t stream) {
  (void)in_sizes; (void)n_in; (void)out_size; (void)ws_size;
  const float* nodes_in  = (const float*)d_in[0];
  const float* edges_in  = (const float*)d_in[1];
  const float* glob_in   = (const float*)d_in[2];
  const int*   senders   = (const int*)d_in[3];
  const int*   receivers = (const int*)d_in[4];

  int pi = 5;
  auto F = [&]() -> const float* { return (const float*)d_in[pi++]; };
  const float* emb_n_w = F(); const float* emb_n_b = F();
  const float* emb_e_w = F(); const float* emb_e_b = F();
  const float* emb_g_w = F(); const float* emb_g_b = F();
  struct StepP {
    const float *ew[4], *eb[4], *aw[3], *ab[3], *nw[3], *nb[3];
    const float *lns_n, *lnb_n, *lns_e, *lnb_e, *lns_g, *lnb_g;
  } st[2];
  for (int s = 0; s < 2; ++s) {
    for (int l = 0; l < 4; ++l) { st[s].ew[l] = F(); st[s].eb[l] = F(); }
    for (int l = 0; l < 3; ++l) { st[s].aw[l] = F(); st[s].ab[l] = F(); }
    for (int l = 0; l < 3; ++l) { st[s].nw[l] = F(); st[s].nb[l] = F(); }
    st[s].lns_n = F(); st[s].lnb_n = F();
    st[s].lns_e = F(); st[s].lnb_e = F();
    st[s].lns_g = F(); st[s].lnb_g = F();
  }
  const float* dec_n_w = F(); const float* dec_n_b = F();
  const float* dec_e_w = F(); const float* dec_e_b = F();

  // workspace carve
  char* wsB = (char*)d_ws;
  size_t off = 0;
  auto carve = [&](size_t bytes) -> void* {
    void* p = wsB + off;
    off += (bytes + 255) & ~(size_t)255;
    return p;
  };
  float* nodes_a = (float*)carve((size_t)N_NODES * 16 * 4);
  float* nodes_b = (float*)carve((size_t)N_NODES * 16 * 4);
  float* edges_a = (float*)carve((size_t)N_EDGES * 128 * 4);
  float* edges_b = (float*)carve((size_t)N_EDGES * 128 * 4);
  float*    sent   = (float*)carve((size_t)N_NODES * 128 * 4);
  float*    recv   = (float*)carve((size_t)N_NODES * 128 * 4);
  float*    denom  = (float*)carve((size_t)N_NODES * 4);
  unsigned* maxenc = (unsigned*)carve((size_t)N_NODES * 4);
  float* logits = (float*)carve((size_t)N_EDGES * 4);
  float* att    = (float*)carve((size_t)N_EDGES * 4);
  float* gbuf   = (float*)carve(64);
  const long zero_n = ((char*)maxenc - (char*)sent) / 4 + N_NODES;  // sent..maxenc

  float* out_nodes = (float*)d_out;                         // [50000,16]
  float* out_edges = out_nodes + (size_t)N_NODES * 16;      // [500000,64]
  float* out_glob  = out_nodes + (size_t)N_NODES * 16 + (size_t)N_EDGES * 64;  // [8]

  const int ETILES16 = N_EDGES / 16;  // 31250
  const int ETILES32 = N_EDGES / 32;  // 15625
  const int NTILES   = N_NODES / 16;  // 3125

  // globals for all steps + final global output
  glob_kernel<<<1, 32, 0, stream>>>(glob_in, emb_g_w, emb_g_b, st[0].lns_g,
                                    st[0].lnb_g, st[1].lns_g, st[1].lnb_g, gbuf,
                                    out_glob);
  // embed
  gemm_16x16_kernel<<<NTILES, 32, 0, stream>>>(nodes_in, emb_n_w, emb_n_b, nodes_a);
  embed_edge_kernel<<<ETILES16, 256, 0, stream>>>(edges_in, emb_e_w, emb_e_b, edges_a);

  float* nodes_cur = nodes_a; float* nodes_next = nodes_b;
  float* edges_cur = edges_a; float* edges_new  = edges_b;

  for (int s = 0; s < 2; ++s) {
    fill_zero_kernel<<<(int)((zero_n + 255) / 256), 256, 0, stream>>>(sent, zero_n);

    EdgeParams ep;
    ep.edges_cur = edges_cur; ep.nodes_cur = nodes_cur; ep.glob = gbuf + 8 * s;
    ep.senders = senders; ep.receivers = receivers;
    ep.ew0 = st[s].ew[0]; ep.eb0 = st[s].eb[0];
    ep.ew1 = st[s].ew[1]; ep.eb1 = st[s].eb[1];
    ep.ew2 = st[s].ew[2]; ep.eb2 = st[s].eb[2];
    ep.ew3 = st[s].ew[3]; ep.eb3 = st[s].eb[3];
    ep.aw0 = st[s].aw[0]; ep.ab0 = st[s].ab[0];
    ep.aw1 = st[s].aw[1]; ep.ab1 = st[s].ab[1];
    ep.aw2 = st[s].aw[2]; ep.ab2 = st[s].ab[2];
    ep.new_edges = edges_new; ep.logits = logits; ep.maxenc = maxenc;
    edge_mlp_kernel<<<ETILES32, 256, 0, stream>>>(ep);

    const int eb1 = (N_EDGES + 255) / 256;
    denom_kernel<<<eb1, 256, 0, stream>>>(logits, receivers, maxenc, denom);
    att_kernel<<<eb1, 256, 0, stream>>>(logits, receivers, maxenc, denom, att);
    scatter_kernel<<<N_EDGES * 32 / 256, 256, 0, stream>>>(edges_new, att, senders,
                                                           receivers, sent, recv);

    NodeParams np;
    np.nodes_cur = nodes_cur; np.sent = sent; np.recv = recv; np.glob = gbuf + 8 * s;
    np.nw0 = st[s].nw[0]; np.nb0 = st[s].nb[0];
    np.nw1 = st[s].nw[1]; np.nb1 = st[s].nb[1];
    np.nw2 = st[s].nw[2]; np.nb2 = st[s].nb[2];
    np.ln_s = st[s].lns_n; np.ln_b = st[s].lnb_n;
    np.nodes_next = nodes_next;
    node_mlp_kernel<<<NTILES, 256, 0, stream>>>(np);

    edge_ln_kernel<<<N_EDGES * 32 / 256, 256, 0, stream>>>(edges_new, edges_cur,
                                                           st[s].lns_e, st[s].lnb_e);

    // swap: edges_new (now LN'd) becomes current; old current is next scratch
    float* t = edges_cur; edges_cur = edges_new; edges_new = t;
    t = nodes_cur; nodes_cur = nodes_next; nodes_next = t;
  }

  // decoder
  gemm_16x16_kernel<<<NTILES, 32, 0, stream>>>(nodes_cur, dec_n_w, dec_n_b, out_nodes);
  dec_edge_kernel<<<ETILES16, 128, 0, stream>>>(edges_cur, dec_e_w, dec_e_b, out_edges);
}